// GetCost_47227460386872
// MI455X (gfx1250) — compile-verified
//
#include <hip/hip_runtime.h>
#include <stdint.h>

// ---------------------------------------------------------------------------
// GetCost (MVS group-wise correlation cost volume) for MI455X / gfx1250.
// Gather-bound: features (105 MB) fit in 192MB L2; one thread per pixel.
//  - ref features staged to LDS via global_load_async_to_lds_b32 (ASYNCcnt),
//    laid out so each thread reads 4 channels per ds_load_b128
//  - x-tap pairs fetched with a single 4B-aligned b64 gather (halves VMEM ops)
//  - boundary/padding folded into precomputed per-(view,depth) pair weights
// ---------------------------------------------------------------------------

constexpr int B_  = 2;
constexpr int V_  = 5;
constexpr int C_  = 32;
constexpr int H_  = 256;
constexpr int W_  = 320;
constexpr int HW  = H_ * W_;
constexpr int G_  = 4;
constexpr int D_  = 4;      // CostNum
constexpr int BLK = 256;    // 8 waves (wave32)
constexpr int NQ  = C_ / 4; // 4-channel chunks

typedef float vfloat2 __attribute__((ext_vector_type(2)));
typedef vfloat2 __attribute__((aligned(4))) vfloat2_u;   // 4B-aligned b64 load
typedef float vfloat4 __attribute__((ext_vector_type(4)));

// ---------------------------------------------------------------------------
// Kernel 1: per (batch, src-view) combined projection  P = src_proj @ inv(ref)
// Both projs are affine (bottom row [0,0,0,1]) -> 3x3 adjugate inverse.
// Writes rot[9], trans[3] per (b, view) into ws.
// ---------------------------------------------------------------------------
__global__ void proj_kernel(const float* __restrict__ pm, float* __restrict__ ws) {
    int t = threadIdx.x;
    if (t >= B_ * (V_ - 1)) return;
    int b = t >> 2;
    int i = t & 3;          // source view index, actual view = i+1

    // make_proj: top3x4 = K[:3,:3] @ E[:3,:4]
    auto build = [&](int v, float A[9], float tv[3]) {
        const float* E = pm + ((size_t)(b * V_ + v) * 2 + 0) * 16;
        const float* K = pm + ((size_t)(b * V_ + v) * 2 + 1) * 16;
        for (int r = 0; r < 3; ++r) {
            for (int c = 0; c < 3; ++c)
                A[r * 3 + c] = K[r * 4 + 0] * E[0 * 4 + c]
                             + K[r * 4 + 1] * E[1 * 4 + c]
                             + K[r * 4 + 2] * E[2 * 4 + c];
            tv[r] = K[r * 4 + 0] * E[3] + K[r * 4 + 1] * E[7] + K[r * 4 + 2] * E[11];
        }
    };

    float Ar[9], tr[3], As[9], ts[3];
    build(0, Ar, tr);
    build(i + 1, As, ts);

    // inverse of Ar (adjugate)
    float c00 = Ar[4] * Ar[8] - Ar[5] * Ar[7];
    float c01 = Ar[5] * Ar[6] - Ar[3] * Ar[8];
    float c02 = Ar[3] * Ar[7] - Ar[4] * Ar[6];
    float det = Ar[0] * c00 + Ar[1] * c01 + Ar[2] * c02;
    float id  = 1.0f / det;
    float Ai[9];
    Ai[0] = c00 * id; Ai[1] = (Ar[2] * Ar[7] - Ar[1] * Ar[8]) * id; Ai[2] = (Ar[1] * Ar[5] - Ar[2] * Ar[4]) * id;
    Ai[3] = c01 * id; Ai[4] = (Ar[0] * Ar[8] - Ar[2] * Ar[6]) * id; Ai[5] = (Ar[2] * Ar[3] - Ar[0] * Ar[5]) * id;
    Ai[6] = c02 * id; Ai[7] = (Ar[1] * Ar[6] - Ar[0] * Ar[7]) * id; Ai[8] = (Ar[0] * Ar[4] - Ar[1] * Ar[3]) * id;

    float ti[3];
    ti[0] = -(Ai[0] * tr[0] + Ai[1] * tr[1] + Ai[2] * tr[2]);
    ti[1] = -(Ai[3] * tr[0] + Ai[4] * tr[1] + Ai[5] * tr[2]);
    ti[2] = -(Ai[6] * tr[0] + Ai[7] * tr[1] + Ai[8] * tr[2]);

    float* o = ws + (size_t)t * 12;
    for (int r = 0; r < 3; ++r) {
        for (int c = 0; c < 3; ++c)
            o[r * 3 + c] = As[r * 3 + 0] * Ai[0 * 3 + c]
                         + As[r * 3 + 1] * Ai[1 * 3 + c]
                         + As[r * 3 + 2] * Ai[2 * 3 + c];
        o[9 + r] = As[r * 3 + 0] * ti[0] + As[r * 3 + 1] * ti[1] + As[r * 3 + 2] * ti[2] + ts[r];
    }
}

// ---------------------------------------------------------------------------
// Kernel 2: cost volume. One thread per output pixel (b,h,w).
// ---------------------------------------------------------------------------
__global__ __launch_bounds__(BLK) void cost_kernel(
    const float* __restrict__ invd,    // (B,1,H,W)
    const float* __restrict__ feats,   // (V,B,C,H,W)
    const float* __restrict__ vwgt,    // (B,V-1,H,W)
    const float* __restrict__ conf,    // (B,H,W)
    const float* __restrict__ p_dint,
    const float* __restrict__ p_dmax,
    const float* __restrict__ p_dmin,
    const float* __restrict__ ws,      // proj data: (B*(V-1)) x 12
    float* __restrict__ out)           // [B*G*D*HW cor] ++ [B*D*HW samples]
{
    // 32 KB. Channel c of thread t lives at ((c>>2)*BLK + t)*4 + (c&3):
    // each thread's 4-channel chunk is 16B contiguous -> ds_load_b128.
    __shared__ float ref_lds[C_ * BLK];

    const int tid = threadIdx.x;
    const int gid = blockIdx.x * BLK + tid;
    const int b   = gid / HW;             // uniform per block (HW % BLK == 0)
    const int p   = gid - b * HW;
    const int h   = p / W_;
    const int w   = p - h * W_;

    // ---- stage ref features (view 0) into LDS asynchronously (ASYNCcnt) ----
    {
        const char* refb  = (const char*)feats + ((size_t)(b * C_) * HW + p) * 4;
        unsigned    lbase = (unsigned)(uintptr_t)(&ref_lds[0]) + (unsigned)(tid * 16);
#pragma unroll
        for (int c = 0; c < C_; ++c) {
            uint64_t ga = (uint64_t)(uintptr_t)(refb + (size_t)c * HW * 4);
            unsigned la = lbase + (unsigned)((c >> 2) * BLK * 16 + (c & 3) * 4);
            asm volatile("global_load_async_to_lds_b32 %0, %1, off"
                         :: "v"(la), "v"(ga) : "memory");
        }
    }

    // ---- depth samples (overlaps with async staging) ----
    const float cur   = invd[gid];
    const float cf    = conf[gid];
    const float dint  = *p_dint;
    const float r0    = (float)(D_ / 2) * dint;
    const float radius = 0.2f * r0 + (1.0f - cf) * 1.8f * r0;   // MIN_R + (1-conf)(MAX_R-MIN_R)
    const float dlo   = cur - radius;
    const float itv   = (2.0f * radius) / (float)(D_ - 1);
    const float min_disp = 1.0f / (*p_dmax);
    const float max_disp = 1.0f / (*p_dmin);

    float dep[D_];
    float* samp = out + (size_t)B_ * G_ * D_ * HW;
#pragma unroll
    for (int d = 0; d < D_; ++d) {
        float s = fminf(fmaxf((float)d * itv + dlo, 0.0f), 1.0f);
        samp[(size_t)(b * D_ + d) * HW + p] = s;
        dep[d] = 1.0f / fmaxf(min_disp + (max_disp - min_disp) * s, 1e-6f);
    }

    // preload all view weights (4 loads in flight during first view's math)
    float wgtv[V_ - 1];
#pragma unroll
    for (int v = 0; v < V_ - 1; ++v)
        wgtv[v] = vwgt[(size_t)(b * (V_ - 1) + v) * HW + p];

    float acc[G_ * D_];
#pragma unroll
    for (int k = 0; k < G_ * D_; ++k) acc[k] = 0.0f;
    float vws = 1e-8f;

    const float fx = (float)w, fy = (float)h;

    // ref LDS data needed from here on (own-lane data only -> no barrier needed)
    asm volatile("s_wait_asynccnt 0" ::: "memory");

    for (int v = 0; v < V_ - 1; ++v) {
        const float* pr  = ws + (size_t)(b * (V_ - 1) + v) * 12;    // uniform
        const float  wgt = wgtv[v];
        vws += wgt;
        const float wg8 = wgt * 0.125f;    // weight * (1/8) for channel-group mean

        const float rx = fmaf(pr[0], fx, fmaf(pr[1], fy, pr[2]));
        const float ry = fmaf(pr[3], fx, fmaf(pr[4], fy, pr[5]));
        const float rz = fmaf(pr[6], fx, fmaf(pr[7], fy, pr[8]));

        // Per depth: two row byte-offsets (b64 pair gathers) + 4 effective
        // pair weights {lo0,hi0,lo1,hi1} with clamp/padding folded in.
        int   roff[D_][2];
        float ewt[D_][4];
#pragma unroll
        for (int d = 0; d < D_; ++d) {
            float Z = fmaf(rz, dep[d], pr[11]);
            Z = (Z == 0.0f) ? 1e-8f : Z;
            const float px = fmaf(rx, dep[d], pr[9])  / Z;
            const float py = fmaf(ry, dep[d], pr[10]) / Z;

            const float x0 = floorf(px), y0 = floorf(py);
            const float wx = px - x0,   wy = py - y0;
            const float x1 = x0 + 1.0f, y1 = y0 + 1.0f;

            const bool vx0 = (x0 >= 0.0f) && (x0 <= (float)(W_ - 1));
            const bool vx1 = (x1 >= 0.0f) && (x1 <= (float)(W_ - 1));
            const bool vy0 = (y0 >= 0.0f) && (y0 <= (float)(H_ - 1));
            const bool vy1 = (y1 >= 0.0f) && (y1 <= (float)(H_ - 1));

            float w00 = (vx0 && vy0) ? (1.0f - wx) * (1.0f - wy) : 0.0f;
            float w10 = (vx1 && vy0) ? wx * (1.0f - wy)          : 0.0f;
            float w01 = (vx0 && vy1) ? (1.0f - wx) * wy          : 0.0f;
            float w11 = (vx1 && vy1) ? wx * wy                   : 0.0f;

            // pair base column: covers elements {xb, xb+1}, always in-bounds
            const int xb  = (int)fminf(fmaxf(x0, 0.0f), (float)(W_ - 2));
            const int yc0 = (int)fminf(fmaxf(y0, 0.0f), (float)(H_ - 1));
            const int yc1 = (int)fminf(fmaxf(y1, 0.0f), (float)(H_ - 1));

            // tap->pair-element routing (x0=W-1 -> tapA uses hi; x0=-1 -> tapB uses lo)
            const bool selAhi = (x0 > (float)(W_ - 2));   // tap x0   -> hi element
            const bool selBhi = (x0 >= 0.0f);             // tap x0+1 -> hi element

            float wl0 = selAhi ? 0.0f : w00, wh0 = selAhi ? w00 : 0.0f;
            if (selBhi) wh0 += w10; else wl0 += w10;
            float wl1 = selAhi ? 0.0f : w01, wh1 = selAhi ? w01 : 0.0f;
            if (selBhi) wh1 += w11; else wl1 += w11;

            ewt[d][0] = wl0; ewt[d][1] = wh0; ewt[d][2] = wl1; ewt[d][3] = wh1;
            roff[d][0] = (yc0 * W_ + xb) * 4;
            roff[d][1] = (yc1 * W_ + xb) * 4;
        }

        // source feature base for (view v+1, batch b): uniform -> SGPR saddr
        const char* srcb = (const char*)feats + (size_t)(((v + 1) * B_ + b) * C_) * HW * 4;

        for (int q = 0; q < NQ; ++q) {            // 4 channels per chunk
            // one ds_load_b128: this thread's 4 ref channels
            const vfloat4 rq = *(const vfloat4*)(ref_lds + (size_t)(q * BLK + tid) * 4);

            // keep gather stream one chunk ahead: global_prefetch_b8
            if (q + 1 < NQ)
                __builtin_prefetch(srcb + (size_t)(q * 4 + 4) * HW * 4 + roff[0][0], 0, 1);

            const int g4 = (q >> 1) * D_;          // channel group = q/2
#pragma unroll
            for (int cc = 0; cc < 4; ++cc) {
                const char* sp = srcb + (size_t)(q * 4 + cc) * HW * 4;
                const float m  = rq[cc] * wg8;
#pragma unroll
                for (int d = 0; d < D_; ++d) {
                    vfloat2 r0p = *(const vfloat2_u*)(sp + roff[d][0]);  // global_load_b64
                    vfloat2 r1p = *(const vfloat2_u*)(sp + roff[d][1]);  // global_load_b64
                    float s = fmaf(r1p.y, ewt[d][3],
                              fmaf(r1p.x, ewt[d][2],
                              fmaf(r0p.y, ewt[d][1],
                                   r0p.x * ewt[d][0])));
                    acc[g4 + d] = fmaf(m, s, acc[g4 + d]);
                }
            }
        }
    }

    const float ivw = 1.0f / vws;
#pragma unroll
    for (int k = 0; k < G_ * D_; ++k)
        out[(size_t)(b * (G_ * D_) + k) * HW + p] = acc[k] * ivw;
}

// ---------------------------------------------------------------------------
extern "C" void kernel_launch(void* const* d_in, const int* in_sizes, int n_in,
                              void* d_out, int out_size, void* d_ws, size_t ws_size,
                              hipStream_t stream) {
    const float* invd  = (const float*)d_in[0];  // inverse_depth
    const float* feats = (const float*)d_in[1];  // features
    const float* pm    = (const float*)d_in[2];  // proj_matrices
    const float* vwgt  = (const float*)d_in[3];  // view_weights
    const float* conf  = (const float*)d_in[4];  // confidence
    const float* dint  = (const float*)d_in[5];  // depth_interval
    const float* dmax  = (const float*)d_in[6];  // depth_max
    const float* dmin  = (const float*)d_in[7];  // depth_min
    // d_in[8] = CostNum (== 4, compile-time specialized)

    float* ws  = (float*)d_ws;    // 8 x 12 floats of projection data
    float* out = (float*)d_out;

    proj_kernel<<<1, 32, 0, stream>>>(pm, ws);

    const int nthreads = B_ * HW;          // 163,840
    cost_kernel<<<nthreads / BLK, BLK, 0, stream>>>(
        invd, feats, vwgt, conf, dint, dmax, dmin, ws, out);
}